// Conv2d_71476845740806
// MI455X (gfx1250) — compile-verified
//
#include <hip/hip_runtime.h>

typedef __attribute__((ext_vector_type(2))) float v2f;
typedef __attribute__((ext_vector_type(4))) float v4f;
typedef __attribute__((ext_vector_type(8))) float v8f;

#define BB 8
#define CC 64
#define HH 256
#define WW 256
#define KK 128
#define OH 128
#define OW 128
// vals spatial: 64 x 64

// ---------------- Kernel 1: batch sum  xsum[c,y,x] = sum_b in[b,c,y,x] -----
__global__ void __launch_bounds__(256) ksum(const float* __restrict__ in,
                                            float* __restrict__ xsum) {
    int t = blockIdx.x * 256 + threadIdx.x;      // 0 .. C*H*W/4 - 1
    const int stride4 = CC * HH * WW / 4;        // per-batch stride in float4
    const v4f* ip = (const v4f*)in;
    v4f acc = ip[t];
#pragma unroll
    for (int b = 1; b < BB; ++b) acc += ip[t + b * stride4];
    ((v4f*)xsum)[t] = acc;
}

// ---------------- Kernel 2: implicit-GEMM conv via V_WMMA_F32_16X16X4_F32 --
// vals[f, a, b] = sum_{c,dy,dx} xsum[c, 2a-1+dy, 2b-1+dx] * w[f,c,dy,dx]
// Workgroup: 256 threads = 8 waves. One m-tile (16 spatial cols of one row a)
// per workgroup; wave w handles filters [16w, 16w+16).
// Reduction index r = c*9 + dy*3 + dx (576 total), chunked 8 channels (72 r).
__global__ void __launch_bounds__(256) kconv(const float* __restrict__ xsum,
                                             const float* __restrict__ wgt,
                                             float* __restrict__ vals) {
    __shared__ float Al[16 * 72];    // [m][r_local], row stride 72
    __shared__ float Bl[36 * 256];   // [r_pair][f*2 + (r&1)] pair-interleaved

    const int tid  = threadIdx.x;
    const int lane = tid & 31;
    const int wave = tid >> 5;           // 0..7
    const int tile = blockIdx.x;         // 0..255
    const int a    = tile >> 2;          // output row 0..63
    const int b0   = (tile & 3) << 4;    // output col base (16-wide)

    const int m    = lane & 15;          // A row / B col within tile
    const int half = lane >> 4;          // K-half selector
    const int fb   = wave << 4;          // filter tile base

    v8f acc = {};

    for (int cc = 0; cc < CC; cc += 8) {
        __syncthreads();
        // ---- stage A: 16 x 72 patch values (zero-padded borders) ----
        for (int e = tid; e < 16 * 72; e += 256) {
            int mm = e / 72, r = e - mm * 72;
            int cl = r / 9, q = r - cl * 9;
            int dy = q / 3, dx = q - dy * 3;
            int y = 2 * a - 1 + dy;
            int x = 2 * (b0 + mm) - 1 + dx;
            float v = 0.f;
            if (y >= 0 && y < HH && x >= 0 && x < WW)
                v = xsum[(cc + cl) * (HH * WW) + y * WW + x];
            Al[e] = v;
        }
        // ---- stage B: 72 x 128 weights, pair-interleaved over r ----
        for (int e = tid; e < 36 * 256; e += 256) {
            int p   = e >> 8;            // r pair 0..35
            int rem = e & 255;
            int f   = rem >> 1;
            int r   = (p << 1) + (rem & 1);
            Bl[e] = wgt[f * 576 + cc * 9 + r];
        }
        __syncthreads();
        // ---- 18 WMMA steps over this chunk ----
#pragma unroll
        for (int rb = 0; rb < 72; rb += 4) {
            int r0 = rb + 2 * half;      // even
            v2f Af = *(const v2f*)&Al[m * 72 + r0];
            v2f Bf = *(const v2f*)&Bl[(r0 >> 1) * 256 + ((fb + m) << 1)];
            acc = __builtin_amdgcn_wmma_f32_16x16x4_f32(
                false, Af, false, Bf, (short)0, acc, false, false);
        }
    }

    // ---- write D: VGPR v -> M = v + 8*half, N = lane&15 ----
    int f    = fb + m;
    int mrow = 8 * half;
    float* dst = &vals[f * 4096 + a * 64 + b0 + mrow];
    v4f lo = {acc[0], acc[1], acc[2], acc[3]};
    v4f hi = {acc[4], acc[5], acc[6], acc[7]};
    *(v4f*)(dst)     = lo;
    *(v4f*)(dst + 4) = hi;
}

// ---------------- Kernel 3: expand vals into the sparse output -------------
__global__ void __launch_bounds__(256) kexpand(const float* __restrict__ vals,
                                               float* __restrict__ out) {
    int t = blockIdx.x * 256 + threadIdx.x;  // float4 index into out
    int jq   = t & 31;        // j0 = 4*jq
    int rest = t >> 5;
    int i    = rest & 127;
    int k    = (rest >> 7) & 127;
    v4f o = {0.f, 0.f, 0.f, 0.f};
    if ((i & 1) == 0) {
        const float* vp = &vals[k * 4096 + (i >> 1) * 64 + (jq << 1)];
        o[0] = vp[0];
        o[2] = vp[1];
    }
    ((v4f*)out)[t] = o;
}

extern "C" void kernel_launch(void* const* d_in, const int* in_sizes, int n_in,
                              void* d_out, int out_size, void* d_ws, size_t ws_size,
                              hipStream_t stream) {
    const float* in  = (const float*)d_in[0];
    const float* wgt = (const float*)d_in[1];
    float* out  = (float*)d_out;
    float* xsum = out;            // reuse first 16 MiB of d_out as scratch
    float* vals = (float*)d_ws;   // 2 MiB

    ksum<<<(CC * HH * WW / 4) / 256, 256, 0, stream>>>(in, xsum);
    kconv<<<256, 256, 0, stream>>>(xsum, wgt, vals);
    kexpand<<<(BB * KK * OH * OW / 4) / 256, 256, 0, stream>>>(vals, out);
}